// MiAMix_38963943309392
// MI455X (gfx1250) — compile-verified
//
#include <hip/hip_runtime.h>

#define N 64
#define C 3
#define H 384
#define W 384
#define K 3
#define HW (H*W)
#define CHW (C*H*W)
#define NCHW (N*C*H*W)

#define TILE 32
#define HALO 3
#define B1W (TILE + 2*HALO)   /* 38 */
#define PITCH 40

typedef __attribute__((ext_vector_type(2))) float v2f;
typedef __attribute__((ext_vector_type(8))) float v8f;

__device__ __forceinline__ int reflect_idx(int r, int n) {
  if (r < 0) r = -r;
  if (r >= n) r = 2*(n-1) - r;
  return r;
}

struct StageParams {
  float lam, mux, muy;
  float cs, sn;
  float x1, x2, y1, y2;   // cutout box
  float inv2sig2;         // gaussian
  int method;
};

// Analytic stage mask value at integer pixel coords (xx = col, yy = row)
__device__ __forceinline__ float stage_val(const StageParams& p, float xx, float yy) {
  if (p.method == 0) return p.lam;
  if (p.method == 1) {
    bool inside = (xx >= p.x1) && (xx < p.x2) && (yy >= p.y1) && (yy < p.y2);
    return inside ? 0.0f : 1.0f;
  }
  float dx = xx - p.mux, dy = yy - p.muy;
  float g = expf(-(dx*dx + dy*dy) * p.inv2sig2);
  return (g > 0.5f) ? 1.0f : 0.0f;
}

__device__ __forceinline__ float corner_val(const StageParams& p, float xf, float yf) {
  bool valid = (xf >= 0.0f) && (xf < (float)W) && (yf >= 0.0f) && (yf < (float)H);
  if (!valid) return 0.0f;
  int xi = (int)fminf(fmaxf(xf, 0.0f), (float)(W-1));
  int yi = (int)fminf(fmaxf(yf, 0.0f), (float)(H-1));
  return stage_val(p, (float)xi, (float)yi);
}

// Rotated (bilinear) stage mask at output pixel (row gy, col gx)
__device__ __forceinline__ float rot_val(const StageParams& p, int gy, int gx) {
  float xn = 2.0f*((float)gx + 0.5f)/(float)W - 1.0f;
  float yn = 2.0f*((float)gy + 0.5f)/(float)H - 1.0f;
  float rx = p.cs*xn - p.sn*yn;
  float ry = p.sn*xn + p.cs*yn;
  float ix = ((rx + 1.0f)*(float)W - 1.0f)*0.5f;
  float iy = ((ry + 1.0f)*(float)H - 1.0f)*0.5f;
  float ix0 = floorf(ix), iy0 = floorf(iy);
  float wx = ix - ix0, wy = iy - iy0;
  float v00 = corner_val(p, ix0,        iy0);
  float v01 = corner_val(p, ix0 + 1.0f, iy0);
  float v10 = corner_val(p, ix0,        iy0 + 1.0f);
  float v11 = corner_val(p, ix0 + 1.0f, iy0 + 1.0f);
  return v00*(1.0f-wx)*(1.0f-wy) + v01*wx*(1.0f-wy)
       + v10*(1.0f-wx)*wy        + v11*wx*wy;
}

// Fused: stage mask -> rotate -> vblur -> hblur -> clip -> merge over K.
// One block per (n, 32x32 tile). All intermediates live in LDS.
__global__ __launch_bounds__(256) void mask_merge_kernel(
    const float* __restrict__ lam, const float* __restrict__ angle,
    const float* __restrict__ sigma, const float* __restrict__ mux,
    const float* __restrict__ muy, const int* __restrict__ method_id,
    const int* __restrict__ ksize, const int* __restrict__ cx,
    const int* __restrict__ cy, const int* __restrict__ stage_valid,
    float* __restrict__ merged)
{
  __shared__ float buf1[B1W*PITCH];   // rotated mask tile + halo (38x38)
  __shared__ float buf2[TILE*PITCH];  // v-blurred (32 rows x 38 cols)

  const int tid = threadIdx.x;
  const int n   = blockIdx.z;
  const int ty0 = blockIdx.y * TILE;
  const int tx0 = blockIdx.x * TILE;

  float acc[4] = {0.f, 0.f, 0.f, 0.f};
  float vsum = 0.f;

  for (int k = 0; k < K; ++k) {
    const int nk = n*K + k;
    const float v = (float)stage_valid[nk];
    vsum += v;

    StageParams p;
    p.lam    = lam[nk];
    p.method = method_id[nk];
    p.mux    = mux[nk];
    p.muy    = muy[nk];
    {
      float rad = angle[nk] * 0.017453292519943295f;
      p.cs = cosf(rad);
      p.sn = sinf(rad);
      float root   = sqrtf(fmaxf(1.0f - p.lam, 0.0f));
      float half_w = floorf(floorf((float)W * root) * 0.5f);
      float half_h = floorf(floorf((float)H * root) * 0.5f);
      float cxf = (float)cx[nk], cyf = (float)cy[nk];
      p.x1 = fmaxf(cxf - half_w, 0.0f);
      p.x2 = fminf(cxf + half_w, (float)W);
      p.y1 = fmaxf(cyf - half_h, 0.0f);
      p.y2 = fminf(cyf + half_h, (float)H);
      float sig = root * ((float)(H > W ? H : W) * 0.25f);
      p.inv2sig2 = 1.0f / (2.0f*sig*sig + 1e-12f);
    }

    // 7-tap Gaussian weights (per n,k), normalized, zeroed outside ksize.
    float kern[7];
    {
      int   half = (ksize[nk] - 1) >> 1;
      float sg   = sigma[nk];
      float ks   = 0.0f;
      #pragma unroll
      for (int i = 0; i < 7; ++i) {
        float t = (float)(i - 3);
        float wgt = (fabsf(t) <= (float)half) ? expf(-0.5f*(t/sg)*(t/sg)) : 0.0f;
        kern[i] = wgt; ks += wgt;
      }
      float inv = 1.0f / ks;
      #pragma unroll
      for (int i = 0; i < 7; ++i) kern[i] *= inv;
    }

    // Fill rotated-mask tile + halo; reflect at image borders (pad happens
    // AFTER rotation in the reference, so reflect output coords directly).
    for (int i = tid; i < B1W*B1W; i += 256) {
      int ly = i / B1W, lx = i % B1W;
      int gy = reflect_idx(ty0 + ly - HALO, H);
      int gx = reflect_idx(tx0 + lx - HALO, W);
      buf1[ly*PITCH + lx] = rot_val(p, gy, gx);
    }
    __syncthreads();

    // Vertical blur (keep horizontal halo columns for the next pass)
    for (int i = tid; i < TILE*B1W; i += 256) {
      int ly = i / B1W, lx = i % B1W;
      float s = 0.0f;
      #pragma unroll
      for (int j = 0; j < 7; ++j) s += kern[j] * buf1[(ly + j)*PITCH + lx];
      buf2[ly*PITCH + lx] = s;
    }
    __syncthreads();

    // Horizontal blur + clip + weighted merge accumulate
    #pragma unroll
    for (int q = 0; q < 4; ++q) {
      int i  = q*256 + tid;
      int ly = i / TILE, lx = i % TILE;
      float s = 0.0f;
      #pragma unroll
      for (int j = 0; j < 7; ++j) s += kern[j] * buf2[ly*PITCH + lx + j];
      s = fminf(fmaxf(s, 0.0f), 1.0f);
      acc[q] += v * s;
    }
    __syncthreads();  // buf2 fully consumed before next k rewrites it
  }

  const float invv = 1.0f / vsum;
  #pragma unroll
  for (int q = 0; q < 4; ++q) {
    int i  = q*256 + tid;
    int ly = i / TILE, lx = i % TILE;
    merged[(size_t)n*HW + (size_t)(ty0 + ly)*W + (tx0 + lx)] = acc[q] * invv;
  }
}

// Spatial mean of merged per image via V_WMMA_F32_16X16X4_F32:
// B = ones(4x16)  =>  D[i][j] = rowsum_i(A) + C[i][j] (replicated across 16 cols).
// Accumulate C over the wave's slice, then sum-all-lanes / 16 = wave total.
__global__ __launch_bounds__(256) void lam_reduce_kernel(
    const float* __restrict__ merged, float* __restrict__ lam_m)
{
  __shared__ float wsum[8];
  const int n    = blockIdx.x;
  const int tid  = threadIdx.x;
  const int wave = tid >> 5;
  const int lane = tid & 31;
  const float* base = merged + (size_t)n*HW + (size_t)wave*(HW/8);
  float total;

#if __has_builtin(__builtin_amdgcn_wmma_f32_16x16x4_f32)
  v8f c = {0.f,0.f,0.f,0.f,0.f,0.f,0.f,0.f};
  v2f b = {1.0f, 1.0f};
  const int iters = (HW/8)/64;  // 288 exactly
  for (int it = 0; it < iters; ++it) {
    v2f a = *(const v2f*)(base + it*64 + lane*2);
    c = __builtin_amdgcn_wmma_f32_16x16x4_f32(
        /*neg_a=*/false, a, /*neg_b=*/false, b,
        /*c_mod=*/(short)0, c, /*reuse_a=*/false, /*reuse_b=*/false);
  }
  float s = c[0]+c[1]+c[2]+c[3]+c[4]+c[5]+c[6]+c[7];
  #pragma unroll
  for (int off = 16; off > 0; off >>= 1) s += __shfl_xor(s, off, 32);
  total = s * (1.0f/16.0f);   // each row-sum replicated across 16 columns
#else
  float s = 0.0f;
  for (int i = lane; i < HW/8; i += 32) s += base[i];
  #pragma unroll
  for (int off = 16; off > 0; off >>= 1) s += __shfl_xor(s, off, 32);
  total = s;
#endif

  if (lane == 0) wsum[wave] = total;
  __syncthreads();
  if (tid == 0) {
    float t = 0.0f;
    #pragma unroll
    for (int i = 0; i < 8; ++i) t += wsum[i];
    lam_m[n] = t / (float)HW;
  }
}

// Final blend, float4 (128-bit) vectorized. Also emits self_flags.
__global__ __launch_bounds__(256) void mix_kernel(
    const float* __restrict__ x, const float* __restrict__ y,
    const int* __restrict__ j_idx, const float* __restrict__ merged,
    const float* __restrict__ lam_m,
    float* __restrict__ out_x, float* __restrict__ out_y,
    float* __restrict__ flags)
{
  const int g = blockIdx.x*256 + threadIdx.x;
  if (g < N) flags[g] = (j_idx[g] == g) ? 1.0f : 0.0f;

  const size_t i4 = (size_t)g * 4;
  if (i4 >= (size_t)NCHW) return;
  const int    n   = (int)(i4 / CHW);
  const size_t rem = i4 % CHW;
  const size_t hw  = rem % HW;   // 4|HW so all 4 elems share n,c and row

  const int   j  = j_idx[n];
  const float lm = lam_m[n];

  const float4 m4  = *(const float4*)(merged + (size_t)n*HW + hw);
  const float4 x4  = *(const float4*)(x + (size_t)n*CHW + rem);
  const float4 xt4 = *(const float4*)(x + (size_t)j*CHW + rem);
  const float4 y4  = *(const float4*)(y + (size_t)n*CHW + rem);
  const float4 yt4 = *(const float4*)(y + (size_t)j*CHW + rem);

  float4 ox, oy;
  ox.x = m4.x*x4.x + (1.0f - m4.x)*xt4.x;
  ox.y = m4.y*x4.y + (1.0f - m4.y)*xt4.y;
  ox.z = m4.z*x4.z + (1.0f - m4.z)*xt4.z;
  ox.w = m4.w*x4.w + (1.0f - m4.w)*xt4.w;
  oy.x = lm*y4.x + (1.0f - lm)*yt4.x;
  oy.y = lm*y4.y + (1.0f - lm)*yt4.y;
  oy.z = lm*y4.z + (1.0f - lm)*yt4.z;
  oy.w = lm*y4.w + (1.0f - lm)*yt4.w;

  *(float4*)(out_x + (size_t)n*CHW + rem) = ox;
  *(float4*)(out_y + (size_t)n*CHW + rem) = oy;
}

extern "C" void kernel_launch(void* const* d_in, const int* in_sizes, int n_in,
                              void* d_out, int out_size, void* d_ws, size_t ws_size,
                              hipStream_t stream)
{
  (void)in_sizes; (void)n_in; (void)out_size; (void)ws_size;
  const float* x           = (const float*)d_in[0];
  const float* y           = (const float*)d_in[1];
  const float* lam         = (const float*)d_in[2];
  const float* angle       = (const float*)d_in[3];
  const float* sigma       = (const float*)d_in[4];
  const float* mux         = (const float*)d_in[5];
  const float* muy         = (const float*)d_in[6];
  const int*   j_idx       = (const int*)d_in[7];
  const int*   method_id   = (const int*)d_in[8];
  const int*   ksize       = (const int*)d_in[9];
  const int*   cx          = (const int*)d_in[10];
  const int*   cy          = (const int*)d_in[11];
  const int*   stage_valid = (const int*)d_in[12];

  float* merged = (float*)d_ws;                 // N*H*W floats (~37.7 MB)
  float* lam_m  = merged + (size_t)N*HW;        // N floats

  float* out_x = (float*)d_out;
  float* out_y = out_x + (size_t)NCHW;
  float* flags = out_y + (size_t)NCHW;

  dim3 g1(W/TILE, H/TILE, N);   // 12 x 12 x 64
  mask_merge_kernel<<<g1, 256, 0, stream>>>(lam, angle, sigma, mux, muy,
                                            method_id, ksize, cx, cy,
                                            stage_valid, merged);

  lam_reduce_kernel<<<N, 256, 0, stream>>>(merged, lam_m);

  const int blocks = (NCHW/4 + 255)/256;
  mix_kernel<<<blocks, 256, 0, stream>>>(x, y, j_idx, merged, lam_m,
                                         out_x, out_y, flags);
}